// YOLO_loss_28750511079897
// MI455X (gfx1250) — compile-verified
//
#include <hip/hip_runtime.h>
#include <hip/hip_bf16.h>

// ---- problem constants (match reference) ----
constexpr int  BATCH   = 32768;
constexpr int  CH      = 17;
constexpr int  GG      = 49;                 // 7x7 cells
constexpr int  PB      = CH * GG;            // 833 floats per batch item
constexpr int  CELLS   = BATCH * GG;         // 1,605,632 cells
constexpr int  THREADS = 256;                // 8 waves (wave32)
constexpr int  NBLK1   = CELLS / THREADS;    // 6272 blocks, exact (no tail)

typedef __attribute__((ext_vector_type(2))) float v2f;
typedef __attribute__((ext_vector_type(8))) float v8f;

__global__ __launch_bounds__(THREADS) void yolo_loss_partial(
    const float* __restrict__ pred, const float* __restrict__ label,
    float* __restrict__ partials)
{
    __shared__ float s_wave[THREADS / 32];

    const int tid  = threadIdx.x;
    const int g    = blockIdx.x * THREADS + tid;   // global cell id
    const int b    = g / GG;                       // batch index
    const int cell = g - b * GG;                   // cell within 7x7
    const int base = b * PB + cell;                // element offset of channel 0

    const float* lp = label + base;
    const float* pp = pred  + base;

    // Channel 0: objectness / prob. d0 = label0 - pred0.
    const float obj = __builtin_nontemporal_load(lp);
    const float p0  = __builtin_nontemporal_load(pp);
    const float d0  = obj - p0;

    // Channels 1..16: weights & sqrt-choice are compile-time (loop fully unrolls).
    float body = 0.f;
    #pragma unroll
    for (int c = 1; c < CH; ++c) {
        const float lv = __builtin_nontemporal_load(lp + c * GG);
        const float ov = __builtin_nontemporal_load(pp + c * GG);
        const float d  = (c == 3 || c == 4)
                           ? (__builtin_sqrtf(lv) - __builtin_sqrtf(ov))
                           : (lv - ov);
        const float w  = (c <= 4) ? 5.0f : 1.0f;   // loc+size weighted, class unweighted
        body += w * d * d;
    }

    // Single select per cell: obj branch vs no-obj branch.
    const float d0sq = d0 * d0;
    float acc = (obj != 0.f) ? (d0sq + body) : (0.5f * d0sq);

    // ---- wave-level reduction on the matrix unit (V_WMMA_F32_16X16X4_F32) ----
    // A[m][0] = acc(lane m), A[m][2] = acc(lane m+16), A[m][1]=A[m][3]=0.
    // With B = ones: D[m][n] = acc[m] + acc[m+16]  (same for every column n).
    // Lane L<16 holds D rows 0..7 in its 8 C VGPRs; lane L>=16 holds rows 8..15.
    v2f a;  a.x = acc; a.y = 0.f;
    v2f bm; bm.x = 1.f; bm.y = 1.f;
    v8f cm = {};
    cm = __builtin_amdgcn_wmma_f32_16x16x4_f32(
            /*neg_a=*/false, a, /*neg_b=*/false, bm,
            /*c_mod=*/(short)0, cm, /*reuse_a=*/false, /*reuse_b=*/false);
    float wsum = cm[0] + cm[1] + cm[2] + cm[3] + cm[4] + cm[5] + cm[6] + cm[7];
    wsum += __shfl_xor(wsum, 16, 32);   // combine rows 0..7 with rows 8..15

    const int wave = tid >> 5;
    const int lane = tid & 31;
    if (lane == 0) s_wave[wave] = wsum;
    __syncthreads();

    if (tid == 0) {
        float s = 0.f;
        #pragma unroll
        for (int w = 0; w < THREADS / 32; ++w) s += s_wave[w];
        partials[blockIdx.x] = s;       // fixed order -> deterministic
    }
}

__global__ __launch_bounds__(THREADS) void yolo_loss_final(
    const float* __restrict__ partials, float* __restrict__ out)
{
    __shared__ double s[THREADS];
    double acc = 0.0;
    for (int i = threadIdx.x; i < NBLK1; i += THREADS)
        acc += (double)partials[i];
    s[threadIdx.x] = acc;
    __syncthreads();
    for (int k = THREADS / 2; k > 0; k >>= 1) {
        if (threadIdx.x < k) s[threadIdx.x] += s[threadIdx.x + k];
        __syncthreads();
    }
    if (threadIdx.x == 0)
        out[0] = (float)(s[0] * (1.0 / ((double)BATCH * (double)GG)));
}

extern "C" void kernel_launch(void* const* d_in, const int* in_sizes, int n_in,
                              void* d_out, int out_size, void* d_ws, size_t ws_size,
                              hipStream_t stream) {
    const float* pred  = (const float*)d_in[0];   // "out"  [B,17,7,7] f32
    const float* label = (const float*)d_in[1];   // "label"[B,17,7,7] f32
    float* res      = (float*)d_out;              // scalar f32
    float* partials = (float*)d_ws;               // NBLK1 floats of scratch

    yolo_loss_partial<<<NBLK1, THREADS, 0, stream>>>(pred, label, partials);
    yolo_loss_final<<<1, THREADS, 0, stream>>>(partials, res);
}